// TreeRCNN_64673617543815
// MI455X (gfx1250) — compile-verified
//
#include <hip/hip_runtime.h>
#include <cstdint>

// Problem constants (match reference)
#define NPTS   65536
#define NANCH  2048
#define MAXP   512
#define MINP   4

// Tiling
#define TILE   2048                 // points per LDS tile
#define NTILE  (NPTS / TILE)        // 32 tiles
#define TILE_ELEMS (TILE * 3)       // 6144 f32 elements = 24576 bytes
#define WAVES  8                    // 8 anchors (waves) per block
#define TPB    (WAVES * 32)         // 256 threads, wave32

typedef unsigned int v4u __attribute__((ext_vector_type(4)));
typedef int          v8i __attribute__((ext_vector_type(8)));
typedef int          v4i __attribute__((ext_vector_type(4)));

// Issue one Tensor-Data-Mover copy: nElems contiguous f32 from gsrc to the LDS address ldsAddr.
// D# built per CDNA5 ISA (CDNA5 ISA): 2-D tensor, one full row as the tile.
__device__ __forceinline__ void tdm_load_tile(const float* gsrc, unsigned ldsAddr)
{
    const unsigned long long ga = (unsigned long long)(uintptr_t)gsrc;
    const unsigned nE = TILE_ELEMS;          // 6144, fits 16-bit tile_dim0

    v4u g0;
    g0.x = 0x1u;                                          // count=1, user descriptor
    g0.y = ldsAddr;                                       // lds_addr (bytes)
    g0.z = (unsigned)(ga & 0xFFFFFFFFu);                  // global_addr[31:0]
    g0.w = (unsigned)((ga >> 32) & 0x01FFFFFFu)           // global_addr[56:32]
         | (2u << 30);                                    // type = 2 ("image")

    v8i g1;
    g1[0] = (int)(2u << 16);                              // wg_mask=0, data_size=2 (4B)
    g1[1] = (int)((nE & 0xFFFFu) << 16);                  // tensor_dim0[15:0]
    g1[2] = (int)((nE >> 16) | (1u << 16));               // tensor_dim0[31:16] | tensor_dim1=1
    g1[3] = (int)((nE & 0xFFFFu) << 16);                  // tensor_dim1 hi=0 | tile_dim0
    g1[4] = 1;                                            // tile_dim1=1, tile_dim2=0
    g1[5] = (int)nE;                                      // tensor_dim0_stride[31:0]
    g1[6] = 0;                                            // stride hi / dim1_stride lo
    g1[7] = 0;

    v4i g2 = {0, 0, 0, 0};                                // dims 2..4 unused
    v4i g3 = {0, 0, 0, 0};

    asm volatile("tensor_load_to_lds %0, %1, %2, %3"
                 :: "s"(g0), "s"(g1), "s"(g2), "s"(g3)
                 : "memory");
}

__global__ __launch_bounds__(TPB)
void roi_gather_kernel(const float* __restrict__ points,
                       const float* __restrict__ anchors,
                       float* __restrict__ out)
{
    // 2 * 24KB point tiles + 16KB index lists + 192B anchors ~= 64.2KB LDS
    __shared__ float tileBuf[2][TILE_ELEMS];
    __shared__ int   sel[WAVES][MAXP];
    __shared__ float anchorBuf[WAVES * 6];

    const int tid  = threadIdx.x;
    const int lane = tid & 31;
    const int wv   = tid >> 5;
    const int a    = blockIdx.x * WAVES + wv;

    // Stage this block's 8 anchors (192B = 12 x b128) into LDS via async loads.
    if (tid < 12) {
        const char*   src = (const char*)(anchors + (size_t)blockIdx.x * (WAVES * 6));
        const unsigned dst = (unsigned)(uintptr_t)(&anchorBuf[0]);
        asm volatile("global_load_async_to_lds_b128 %0, %1, off"
                     :: "v"(dst + (unsigned)(tid * 16)), "v"(src + tid * 16)
                     : "memory");
    }
    // Kick off tile 0 through the Tensor Data Mover (one descriptor, wave 0).
    if (wv == 0) {
        tdm_load_tile(points, (unsigned)(uintptr_t)(&tileBuf[0][0]));
    }

    // Anchors visible to everyone (tile 0 gated later by TENSORcnt).
    asm volatile("s_wait_asynccnt 0" ::: "memory");
    __syncthreads();

    const float cx = anchorBuf[wv * 6 + 0];
    const float cy = anchorBuf[wv * 6 + 1];
    const float w  = anchorBuf[wv * 6 + 3];
    const float l  = anchorBuf[wv * 6 + 4];
    const float h  = anchorBuf[wv * 6 + 5];
    const float xmin = cx - 0.5f * w, xmax = cx + 0.5f * w;
    const float ymin = cy - 0.5f * l, ymax = cy + 0.5f * l;

    int count = 0;  // total in-box points for this anchor (wave-uniform)
    for (int t = 0; t < NTILE; ++t) {
        // All waves drain TENSORcnt (safe even if a wave issued nothing), then
        // barrier: tile t is in LDS for everyone, and everyone finished reading
        // the buffer the next TDM descriptor will overwrite.
        __builtin_amdgcn_s_wait_tensorcnt(0);
        __syncthreads();
        if (wv == 0 && t + 1 < NTILE) {
            tdm_load_tile(points + (size_t)(t + 1) * TILE_ELEMS,
                          (unsigned)(uintptr_t)(&tileBuf[(t + 1) & 1][0]));
        }

        const float* tb = &tileBuf[t & 1][0];
        #pragma unroll 4
        for (int c = 0; c < TILE / 32; ++c) {
            const int   p  = c * 32 + lane;
            const float px = tb[p * 3 + 0];
            const float py = tb[p * 3 + 1];
            const float pz = tb[p * 3 + 2];
            const bool inside = (px >= xmin) && (px <= xmax) &&
                                (py >= ymin) && (py <= ymax) &&
                                (pz >= 0.0f) && (pz <= h);
            // wave32 ballot -> ordered compaction ranks
            const unsigned m = (unsigned)__ballot(inside);
            if (inside) {
                const int r = count + __popc(m & ((1u << lane) - 1u));
                if (r < MAXP) sel[wv][r] = t * TILE + p;
            }
            count += __popc(m);
        }
    }

    // Emit: 512 points per anchor, cycled modulo count, recentered; zero if count < MIN_PTS.
    float* oBase = out + (size_t)a * (MAXP * 3);
    if (count >= MINP) {
        for (int j = lane; j < MAXP; j += 32) {
            // for count >= MAXP: j % count == j (j < 512 <= count)
            const int slot = (count >= MAXP) ? j : (j % count);
            const int idx  = sel[wv][slot];
            oBase[j * 3 + 0] = points[idx * 3 + 0] - cx;
            oBase[j * 3 + 1] = points[idx * 3 + 1] - cy;
            oBase[j * 3 + 2] = points[idx * 3 + 2];
        }
    } else {
        for (int j = lane; j < MAXP; j += 32) {
            oBase[j * 3 + 0] = 0.0f;
            oBase[j * 3 + 1] = 0.0f;
            oBase[j * 3 + 2] = 0.0f;
        }
    }
}

extern "C" void kernel_launch(void* const* d_in, const int* in_sizes, int n_in,
                              void* d_out, int out_size, void* d_ws, size_t ws_size,
                              hipStream_t stream) {
    const float* points  = (const float*)d_in[0];   // 65536 x 3 f32
    const float* anchors = (const float*)d_in[1];   // 2048 x 6 f32
    (void)in_sizes; (void)n_in; (void)out_size; (void)d_ws; (void)ws_size;
    float* out = (float*)d_out;                     // 2048 x 512 x 3 f32

    dim3 grid(NANCH / WAVES);
    dim3 block(TPB);
    hipLaunchKernelGGL(roi_gather_kernel, grid, block, 0, stream,
                       points, anchors, out);
}